// SequentialTwoStagePredictor_704374636751
// MI455X (gfx1250) — compile-verified
//
#include <hip/hip_runtime.h>

// ---------------- problem constants (from reference) ----------------
#define NROWS   100000
#define NEDGES  1600000
#define ETOT    (NEDGES + NROWS)     // edges + self loops
#define IN_F    64
#define HID     32
#define HEADS   2
#define OUT_CH  24
#define NEG_SLOPE 0.2f

typedef float v2f __attribute__((ext_vector_type(2)));
typedef float v8f __attribute__((ext_vector_type(8)));

#define WMMA_F32(a, b, c) \
  __builtin_amdgcn_wmma_f32_16x16x4_f32(false, (a), false, (b), (short)0, (c), false, false)

// ---------------- helpers ----------------
__device__ __forceinline__ unsigned ordEnc(float f) {
  unsigned u = __float_as_uint(f);
  return (u & 0x80000000u) ? ~u : (u | 0x80000000u);
}
__device__ __forceinline__ float ordDec(unsigned o) {
  unsigned u = (o & 0x80000000u) ? (o & 0x7FFFFFFFu) : ~o;
  return __uint_as_float(u);
}
#define ORD_NEG_INF 0x007FFFFFu   // ordEnc(-inf)

// silu + 5 quadratic B-spline bases on the fixed grid (GRID=3, ORDER=2, range [-1,1])
__device__ __forceinline__ void kan_expand(float x, float* out6) {
  out6[0] = x / (1.0f + __expf(-x));          // silu
  const float h = 2.0f / 3.0f;
  float g[8];
#pragma unroll
  for (int j = 0; j < 8; ++j) g[j] = -1.0f + (float)(j - 2) * h;
  float b0[7];
#pragma unroll
  for (int j = 0; j < 7; ++j) b0[j] = (x >= g[j] && x < g[j + 1]) ? 1.0f : 0.0f;
  float b1[6];
#pragma unroll
  for (int j = 0; j < 6; ++j)
    b1[j] = (x - g[j]) * 1.5f * b0[j] + (g[j + 2] - x) * 1.5f * b0[j + 1];
#pragma unroll
  for (int j = 0; j < 5; ++j)
    out6[1 + j] = (x - g[j]) * 0.75f * b1[j] + (g[j + 3] - x) * 0.75f * b1[j + 1];
}

// ---------------- GAT input projection: H[N,64] = X[N,64] @ W[64,64] ----------------
__global__ __launch_bounds__(256) void gemm_x_w(const float* __restrict__ X,
                                                const float* __restrict__ W,
                                                float* __restrict__ H) {
  __shared__ float xs[128][65];
  __shared__ float wsl[64][65];
  const int row0 = blockIdx.x * 128;
  const int tid  = threadIdx.x;
  for (int i = tid; i < 64 * 64; i += 256) wsl[i >> 6][i & 63] = W[i];
  for (int i = tid; i < 128 * 64; i += 256) {
    int r = i >> 6, cx = i & 63, gr = row0 + r;
    xs[r][cx] = (gr < NROWS) ? X[gr * 64 + cx] : 0.0f;
  }
  __syncthreads();
  const int wave = tid >> 5, lane = tid & 31;
  const int mrow = lane & 15, kh = lane >> 4;
  v8f acc[4];
#pragma unroll
  for (int t = 0; t < 4; ++t)
#pragma unroll
    for (int j = 0; j < 8; ++j) acc[t][j] = 0.0f;

#pragma unroll
  for (int s = 0; s < 16; ++s) {
    const int kk = s * 4;
    v2f a;
    a.x = xs[wave * 16 + mrow][kk + kh * 2 + 0];
    a.y = xs[wave * 16 + mrow][kk + kh * 2 + 1];
#pragma unroll
    for (int t = 0; t < 4; ++t) {
      v2f b;
      b.x = wsl[kk + kh * 2 + 0][t * 16 + mrow];
      b.y = wsl[kk + kh * 2 + 1][t * 16 + mrow];
      acc[t] = WMMA_F32(a, b, acc[t]);
    }
  }
#pragma unroll
  for (int t = 0; t < 4; ++t) {
    const int col = t * 16 + mrow;
#pragma unroll
    for (int j = 0; j < 8; ++j) {
      const int r = row0 + wave * 16 + kh * 8 + j;
      if (r < NROWS) H[r * 64 + col] = acc[t][j];
    }
  }
}

// ---------------- per-node attention coefficients + init ----------------
__global__ void k_att(const float* __restrict__ h, const float* __restrict__ attS,
                      const float* __restrict__ attD, float* __restrict__ asrc,
                      float* __restrict__ adst, unsigned* __restrict__ m,
                      float* __restrict__ denom) {
  int n = blockIdx.x * blockDim.x + threadIdx.x;
  if (n >= NROWS) return;
#pragma unroll
  for (int hd = 0; hd < HEADS; ++hd) {
    float s = 0.f, sd = 0.f;
#pragma unroll
    for (int d = 0; d < HID; ++d) {
      float hv = h[n * 64 + hd * HID + d];
      s  += hv * attS[hd * HID + d];
      sd += hv * attD[hd * HID + d];
    }
    asrc[n * 2 + hd]  = s;
    adst[n * 2 + hd]  = sd;
    m[n * 2 + hd]     = ORD_NEG_INF;
    denom[n * 2 + hd] = 0.0f;
  }
}

__global__ void k_zero(float* __restrict__ p, int count) {
  int t = blockIdx.x * blockDim.x + threadIdx.x;
  if (t < count) p[t] = 0.0f;
}

// ---------------- edge pass 1: segment max via ordered-uint atomic max ----------------
__device__ __forceinline__ void edge_sd(const int* __restrict__ ei, int t, int& src, int& dst) {
  if (t < NEDGES) { src = ei[t]; dst = ei[NEDGES + t]; }
  else            { src = dst = t - NEDGES; }
}

__global__ void k_edge_max(const int* __restrict__ ei, const float* __restrict__ asrc,
                           const float* __restrict__ adst, unsigned* __restrict__ m) {
  int t = blockIdx.x * blockDim.x + threadIdx.x;
  if (t >= ETOT) return;
  int src, dst; edge_sd(ei, t, src, dst);
#pragma unroll
  for (int hd = 0; hd < HEADS; ++hd) {
    float v = asrc[src * 2 + hd] + adst[dst * 2 + hd];
    v = v > 0.f ? v : NEG_SLOPE * v;
    atomicMax(&m[dst * 2 + hd], ordEnc(v));
  }
}

// ---------------- edge pass 2: exp + segment sum ----------------
__global__ void k_edge_exp(const int* __restrict__ ei, const float* __restrict__ asrc,
                           const float* __restrict__ adst, const unsigned* __restrict__ m,
                           float* __restrict__ denom, float* __restrict__ estore) {
  int t = blockIdx.x * blockDim.x + threadIdx.x;
  if (t >= ETOT) return;
  int src, dst; edge_sd(ei, t, src, dst);
#pragma unroll
  for (int hd = 0; hd < HEADS; ++hd) {
    float v = asrc[src * 2 + hd] + adst[dst * 2 + hd];
    v = v > 0.f ? v : NEG_SLOPE * v;
    float e = __expf(v - ordDec(m[dst * 2 + hd]));
    estore[t * 2 + hd] = e;
    atomicAdd(&denom[dst * 2 + hd], e);
  }
}

// ---------------- edge pass 3: weighted message aggregation ----------------
__global__ void k_edge_agg(const int* __restrict__ ei, const float* __restrict__ h,
                           const float* __restrict__ estore, const float* __restrict__ denom,
                           float* __restrict__ acc) {
  int tid = blockIdx.x * blockDim.x + threadIdx.x;
  int t = tid >> 2;
  if (t >= ETOT) return;
  int hd = (tid >> 1) & 1, half = tid & 1;
  int src, dst; edge_sd(ei, t, src, dst);
  float alpha = estore[t * 2 + hd] / (denom[dst * 2 + hd] + 1e-16f);
  int base = hd * HID + half * 16;
  const float4* hp = (const float4*)&h[src * 64 + base];
#pragma unroll
  for (int q = 0; q < 4; ++q) {
    float4 hv = hp[q];
    atomicAdd(&acc[dst * 64 + base + q * 4 + 0], hv.x * alpha);
    atomicAdd(&acc[dst * 64 + base + q * 4 + 1], hv.y * alpha);
    atomicAdd(&acc[dst * 64 + base + q * 4 + 2], hv.z * alpha);
    atomicAdd(&acc[dst * 64 + base + q * 4 + 3], hv.w * alpha);
  }
}

// ---------------- head mean + bias -> concat buffer column block g ----------------
__global__ void k_finalize(const float* __restrict__ acc, const float* __restrict__ bias,
                           float* __restrict__ X0, int g) {
  int t = blockIdx.x * blockDim.x + threadIdx.x;
  if (t >= NROWS * HID) return;
  int n = t >> 5, d = t & 31;
  X0[n * 99 + g * HID + d] = 0.5f * (acc[n * 64 + d] + acc[n * 64 + HID + d]) + bias[d];
}

__global__ void k_alphas(const float* __restrict__ alphas, float* __restrict__ wsm) {
  float a0 = alphas[0], a1 = alphas[1], a2 = alphas[2];
  float mx = fmaxf(a0, fmaxf(a1, a2));
  float e0 = __expf(a0 - mx), e1 = __expf(a1 - mx), e2 = __expf(a2 - mx);
  float s = e0 + e1 + e2;
  wsm[0] = e0 / s; wsm[1] = e1 / s; wsm[2] = e2 / s;
}

__global__ void k_scalars(const float* __restrict__ wsm, float* __restrict__ X0) {
  int t = blockIdx.x * blockDim.x + threadIdx.x;
  if (t >= NROWS * 3) return;
  int n = t / 3, j = t % 3;
  X0[n * 99 + 96 + j] = wsm[j];
}

// ---------------- fuse KAN weights: Wc[o, f*6+0]=base_w, [o,f*6+1+k]=spline_w*scaler ----
__global__ void k_prep_wc(const float* __restrict__ base_w, const float* __restrict__ spline_w,
                          const float* __restrict__ scaler, int in_f, int out_f, int outpad,
                          int Kpad, float* __restrict__ Wc) {
  int t = blockIdx.x * blockDim.x + threadIdx.x;
  if (t >= outpad * Kpad) return;
  int o = t / Kpad, cg = t % Kpad;
  float v = 0.0f;
  if (o < out_f && cg < 6 * in_f) {
    int f = cg / 6, slot = cg % 6, idx = o * in_f + f;
    v = (slot == 0) ? base_w[idx] : spline_w[idx * 5 + (slot - 1)] * scaler[idx];
  }
  Wc[t] = v;
}

// ---------------- fused KAN linear: expand(silu+bsplines) in LDS, WMMA f32 GEMM ----------
template <int NT>
__global__ __launch_bounds__(256) void kan_gemm(const float* __restrict__ Xin, int in_f,
                                                int nchunks, const float* __restrict__ Wc,
                                                int Kpad, int out_f, float* __restrict__ Y,
                                                int ldy, int act) {
  __shared__ float uLds[128][49];
  __shared__ float wLds[NT * 16][49];
  const int row0 = blockIdx.x * 128;
  const int tid  = threadIdx.x;
  const int wave = tid >> 5, lane = tid & 31;
  const int mrow = lane & 15, kh = lane >> 4;
  const int outpad = NT * 16;

  v8f acc[NT];
#pragma unroll
  for (int t = 0; t < NT; ++t)
#pragma unroll
    for (int j = 0; j < 8; ++j) acc[t][j] = 0.0f;

  for (int c = 0; c < nchunks; ++c) {
    // stage fused-weight chunk [outpad x 48]
    for (int i = tid; i < outpad * 48; i += 256) {
      int o = i / 48, j = i % 48;
      wLds[o][j] = Wc[o * Kpad + c * 48 + j];
    }
    // stage expanded input chunk [128 rows x 8 features x 6 slots]
    for (int i = tid; i < 128 * 8; i += 256) {
      int r = i >> 3, fi = i & 7;
      int f = c * 8 + fi, gr = row0 + r;
      float e6[6] = {0.f, 0.f, 0.f, 0.f, 0.f, 0.f};
      if (f < in_f && gr < NROWS) kan_expand(Xin[gr * in_f + f], e6);
#pragma unroll
      for (int s = 0; s < 6; ++s) uLds[r][fi * 6 + s] = e6[s];
    }
    __syncthreads();
#pragma unroll
    for (int s = 0; s < 12; ++s) {
      const int kk = s * 4;
      v2f a;
      a.x = uLds[wave * 16 + mrow][kk + kh * 2 + 0];
      a.y = uLds[wave * 16 + mrow][kk + kh * 2 + 1];
#pragma unroll
      for (int t = 0; t < NT; ++t) {
        v2f b;
        b.x = wLds[t * 16 + mrow][kk + kh * 2 + 0];
        b.y = wLds[t * 16 + mrow][kk + kh * 2 + 1];
        acc[t] = WMMA_F32(a, b, acc[t]);
      }
    }
    __syncthreads();
  }
#pragma unroll
  for (int t = 0; t < NT; ++t) {
    const int col = t * 16 + mrow;
#pragma unroll
    for (int j = 0; j < 8; ++j) {
      const int r = row0 + wave * 16 + kh * 8 + j;
      if (col < out_f && r < NROWS) {
        float v = acc[t][j];
        if (act) v = fmaxf(v, 0.0f);
        Y[(long)r * ldy + col] = v;
      }
    }
  }
}

// ---------------- host launcher ----------------
extern "C" void kernel_launch(void* const* d_in, const int* in_sizes, int n_in,
                              void* d_out, int out_size, void* d_ws, size_t ws_size,
                              hipStream_t stream) {
  (void)in_sizes; (void)n_in; (void)out_size; (void)ws_size;

  // inputs flattened in setup_inputs() insertion order
  const float* x = (const float*)d_in[0];
  const int* ei[3] = {(const int*)d_in[1], (const int*)d_in[2], (const int*)d_in[3]};
  const float* alphas = (const float*)d_in[4];
  const float *gW[3], *gAS[3], *gAD[3], *gB[3];
  for (int g = 0; g < 3; ++g) {
    gW[g]  = (const float*)d_in[5 + g * 4 + 0];
    gAS[g] = (const float*)d_in[5 + g * 4 + 1];
    gAD[g] = (const float*)d_in[5 + g * 4 + 2];
    gB[g]  = (const float*)d_in[5 + g * 4 + 3];
  }
  // 6 KAN layers: fusion[0..1], classifier[0..1], regressor[0..1]
  const float *kBW[6], *kSW[6], *kSC[6];
  for (int l = 0; l < 6; ++l) {
    kBW[l] = (const float*)d_in[17 + l * 4 + 0];
    kSW[l] = (const float*)d_in[17 + l * 4 + 1];
    kSC[l] = (const float*)d_in[17 + l * 4 + 2];
  }
  static const int in_f[6]   = {99, 32, 32, 16, 32, 16};
  static const int out_f[6]  = {32, 32, 16, 1, 16, 24};
  static const int outpad[6] = {32, 32, 16, 16, 16, 32};
  static const int chunks[6] = {13, 4, 4, 2, 4, 2};

  // workspace carve-up
  char* base = (char*)d_ws;
  size_t off = 0;
  auto alloc = [&](size_t bytes) -> void* {
    void* p = base + off;
    off = (off + bytes + 255) & ~(size_t)255;
    return p;
  };
  float*    hbuf   = (float*)alloc((size_t)NROWS * 64 * 4);
  float*    accb   = (float*)alloc((size_t)NROWS * 64 * 4);
  float*    asrc   = (float*)alloc((size_t)NROWS * 2 * 4);
  float*    adst   = (float*)alloc((size_t)NROWS * 2 * 4);
  unsigned* mbuf   = (unsigned*)alloc((size_t)NROWS * 2 * 4);
  float*    denom  = (float*)alloc((size_t)NROWS * 2 * 4);
  float*    estore = (float*)alloc((size_t)ETOT * 2 * 4);
  float*    X0     = (float*)alloc((size_t)NROWS * 99 * 4);
  float*    F1     = (float*)alloc((size_t)NROWS * 32 * 4);
  float*    FUSED  = (float*)alloc((size_t)NROWS * 32 * 4);
  float*    C1     = (float*)alloc((size_t)NROWS * 16 * 4);
  float*    R1     = (float*)alloc((size_t)NROWS * 16 * 4);
  float*    wsm    = (float*)alloc(3 * 4);
  float*    Wc[6];
  for (int l = 0; l < 6; ++l) Wc[l] = (float*)alloc((size_t)outpad[l] * chunks[l] * 48 * 4);

  const dim3 blk(256);
  const int gN128 = (NROWS + 127) / 128;
  const int gN    = (NROWS + 255) / 256;
  const int gN64  = (NROWS * 64 + 255) / 256;
  const int gN32  = (NROWS * 32 + 255) / 256;
  const int gN3   = (NROWS * 3 + 255) / 256;
  const int gE    = (ETOT + 255) / 256;
  const int gE4   = (ETOT * 4 + 255) / 256;

  // ---- three GAT branches (sequential, shared scratch) ----
  for (int g = 0; g < 3; ++g) {
    gemm_x_w<<<gN128, blk, 0, stream>>>(x, gW[g], hbuf);
    k_att<<<gN, blk, 0, stream>>>(hbuf, gAS[g], gAD[g], asrc, adst, mbuf, denom);
    k_zero<<<gN64, blk, 0, stream>>>(accb, NROWS * 64);
    k_edge_max<<<gE, blk, 0, stream>>>(ei[g], asrc, adst, mbuf);
    k_edge_exp<<<gE, blk, 0, stream>>>(ei[g], asrc, adst, mbuf, denom, estore);
    k_edge_agg<<<gE4, blk, 0, stream>>>(ei[g], hbuf, estore, denom, accb);
    k_finalize<<<gN32, blk, 0, stream>>>(accb, gB[g], X0, g);
  }
  k_alphas<<<1, 1, 0, stream>>>(alphas, wsm);
  k_scalars<<<gN3, blk, 0, stream>>>(wsm, X0);

  // ---- fuse KAN weights ----
  for (int l = 0; l < 6; ++l) {
    int total = outpad[l] * chunks[l] * 48;
    k_prep_wc<<<(total + 255) / 256, blk, 0, stream>>>(kBW[l], kSW[l], kSC[l], in_f[l],
                                                       out_f[l], outpad[l], chunks[l] * 48,
                                                       Wc[l]);
  }

  float* out = (float*)d_out;  // [N logits ; N x 24 regression]
  // fusion
  kan_gemm<2><<<gN128, blk, 0, stream>>>(X0, 99, 13, Wc[0], 13 * 48, 32, F1, 32, 0);
  kan_gemm<2><<<gN128, blk, 0, stream>>>(F1, 32, 4, Wc[1], 4 * 48, 32, FUSED, 32, 1);
  // classifier
  kan_gemm<1><<<gN128, blk, 0, stream>>>(FUSED, 32, 4, Wc[2], 4 * 48, 16, C1, 16, 0);
  kan_gemm<1><<<gN128, blk, 0, stream>>>(C1, 16, 2, Wc[3], 2 * 48, 1, out, 1, 0);
  // regressor
  kan_gemm<1><<<gN128, blk, 0, stream>>>(FUSED, 32, 4, Wc[4], 4 * 48, 16, R1, 16, 0);
  kan_gemm<2><<<gN128, blk, 0, stream>>>(R1, 16, 2, Wc[5], 2 * 48, 24, out + NROWS, 24, 0);
}